// Temporal_Attention_layer_1580547968429
// MI455X (gfx1250) — compile-verified
//
#include <hip/hip_runtime.h>
#include <hip/hip_bf16.h>

// B=32, N=512, F=64, T=128
#define B_ 32
#define N_ 512
#define F_ 64
#define T_ 128

typedef __attribute__((ext_vector_type(2))) float v2f;
typedef __attribute__((ext_vector_type(4))) float v4f;
typedef __attribute__((ext_vector_type(8))) float v8f;

// ---------------------------------------------------------------------------
// Kernel 0: zero the lhs_pre accumulator (atomicAdd target)
// ---------------------------------------------------------------------------
__global__ __launch_bounds__(256) void k0_zero(float* __restrict__ p, int n) {
  int i = blockIdx.x * 256 + threadIdx.x;
  if (i < n) p[i] = 0.0f;
}

// ---------------------------------------------------------------------------
// Kernel 1: single streaming pass over x (512 MB), producing BOTH
//   rhs[b,n,t]     = sum_f U3[f] * x[b,n,f,t]
//   lhs_pre[b,t,f] = sum_n U1[n] * x[b,n,f,t]   (via fp32 global atomics)
// x is read exactly once and exceeds the 192 MB L2 -> non-temporal loads so
// the streamed data does not evict the reused intermediates from L2.
// ---------------------------------------------------------------------------
__global__ __launch_bounds__(256) void k1_stream(
    const float* __restrict__ x, const float* __restrict__ U1,
    const float* __restrict__ U3, float* __restrict__ lhs_pre,
    float* __restrict__ rhs) {
  const int b   = blockIdx.y;
  const int n0  = blockIdx.x * 32;
  const int tid = threadIdx.x;
  const int t4  = tid & 31;
  const int fg  = tid >> 5;

  __shared__ float4 red[8][32];

  float u3[8];
#pragma unroll
  for (int j = 0; j < 8; ++j) u3[j] = U3[fg * 8 + j];

  float lacc[8][4];
#pragma unroll
  for (int j = 0; j < 8; ++j)
#pragma unroll
    for (int c = 0; c < 4; ++c) lacc[j][c] = 0.0f;

  for (int i = 0; i < 32; ++i) {
    const int n = n0 + i;
    const float u1 = U1[n];                       // uniform -> s_load
    const float* xp = x + ((size_t)b * N_ + n) * (F_ * T_);
    float r0 = 0.f, r1 = 0.f, r2 = 0.f, r3 = 0.f;
#pragma unroll
    for (int j = 0; j < 8; ++j) {
      // non-temporal b128 load: stream x through without polluting L2
      const v4f v =
          __builtin_nontemporal_load((const v4f*)(xp + (fg * 8 + j) * T_ + t4 * 4));
      r0 += v.x * u3[j]; r1 += v.y * u3[j]; r2 += v.z * u3[j]; r3 += v.w * u3[j];
      lacc[j][0] += v.x * u1; lacc[j][1] += v.y * u1;
      lacc[j][2] += v.z * u1; lacc[j][3] += v.w * u1;
    }
    red[fg][t4] = make_float4(r0, r1, r2, r3);
    __syncthreads();
    if (fg == 0) {                                // wave 0 reduces 8 groups
      float4 s = red[0][t4];
#pragma unroll
      for (int k = 1; k < 8; ++k) {
        float4 v = red[k][t4];
        s.x += v.x; s.y += v.y; s.z += v.z; s.w += v.w;
      }
      *(float4*)(rhs + ((size_t)b * N_ + n) * T_ + t4 * 4) = s;  // RT: reused by k3
    }
    __syncthreads();
  }

  // flush lhs_pre partials: 32 fp32 atomics per thread, once per block
#pragma unroll
  for (int j = 0; j < 8; ++j) {
    const int f = fg * 8 + j;
#pragma unroll
    for (int c = 0; c < 4; ++c) {
      const int t = t4 * 4 + c;
      unsafeAtomicAdd(&lhs_pre[((size_t)b * T_ + t) * F_ + f], lacc[j][c]);
    }
  }
}

// ---------------------------------------------------------------------------
// fp32 WMMA helper: one 16x16 output tile, K = 4*ksteps.
// A: row-major MxK (lda), B: row-major KxN tile origin (ldb).
// A frag layout (ISA 16x4 f32): v0 = A[m][k0+2*half], v1 = A[m][k0+2*half+1]
// B frag layout: v0 = B[k0+2*half][n], v1 = B[k0+2*half+1][n]
// ---------------------------------------------------------------------------
__device__ __forceinline__ v8f wmma_tile_f32(const float* __restrict__ A, int lda,
                                             const float* __restrict__ Bm, int ldb,
                                             int ksteps, int lm, int half) {
  v8f c = {};
  for (int kk = 0; kk < ksteps; ++kk) {
    const int k0 = kk * 4 + 2 * half;
    v2f a, bb;
    a.x  = A[lm * lda + k0];
    a.y  = A[lm * lda + k0 + 1];
    bb.x = Bm[k0 * ldb + lm];
    bb.y = Bm[(k0 + 1) * ldb + lm];
    c = __builtin_amdgcn_wmma_f32_16x16x4_f32(false, a, false, bb,
                                              (short)0, c, false, false);
  }
  return c;
}

// D layout: VGPR i -> row (i + 8*half), col lm
__device__ __forceinline__ void store_tile(float* __restrict__ D, int ldd,
                                           v8f c, int lm, int half) {
#pragma unroll
  for (int i = 0; i < 8; ++i) D[(i + 8 * half) * ldd + lm] = c[i];
}

// ---------------------------------------------------------------------------
// Kernel 2: lhs[b,t,n] = lhs_pre[b,t,:] @ U2   ((B*T)x64 @ 64x512)
// ---------------------------------------------------------------------------
__global__ __launch_bounds__(256) void k2_lhs(const float* __restrict__ lhs_pre,
                                              const float* __restrict__ U2,
                                              float* __restrict__ lhs) {
  const int wid    = blockIdx.x * 8 + (threadIdx.x >> 5);
  const int tile_m = wid >> 5;            // 0..255  (M = 4096/16)
  const int tile_n = wid & 31;            // 0..31   (N = 512/16)
  const int lane = threadIdx.x & 31;
  const int half = lane >> 4, lm = lane & 15;
  v8f c = wmma_tile_f32(lhs_pre + (size_t)tile_m * 16 * F_, F_,
                        U2 + tile_n * 16, N_, F_ / 4, lm, half);
  store_tile(lhs + (size_t)tile_m * 16 * N_ + tile_n * 16, N_, c, lm, half);
}

// ---------------------------------------------------------------------------
// Kernel 3: product[b] = lhs[b] (128x512) @ rhs[b] (512x128)
// ---------------------------------------------------------------------------
__global__ __launch_bounds__(256) void k3_prod(const float* __restrict__ lhs,
                                               const float* __restrict__ rhs,
                                               float* __restrict__ prod) {
  const int wid    = blockIdx.x * 8 + (threadIdx.x >> 5);   // 0..2047
  const int b      = wid >> 6;
  const int tile_m = (wid >> 3) & 7;
  const int tile_n = wid & 7;
  const int lane = threadIdx.x & 31;
  const int half = lane >> 4, lm = lane & 15;
  v8f c = wmma_tile_f32(lhs + (size_t)b * T_ * N_ + (size_t)tile_m * 16 * N_, N_,
                        rhs + (size_t)b * N_ * T_ + tile_n * 16, T_,
                        N_ / 4, lm, half);
  store_tile(prod + (size_t)b * T_ * T_ + (size_t)tile_m * 16 * T_ + tile_n * 16,
             T_, c, lm, half);
}

// ---------------------------------------------------------------------------
// Kernel 4: sig = sigmoid(product + be), in place (be broadcast over b)
// ---------------------------------------------------------------------------
__global__ __launch_bounds__(256) void k4_sig(float* __restrict__ prod,
                                              const float* __restrict__ be) {
  const int idx = blockIdx.x * 256 + threadIdx.x;       // < B*T*T
  const float v = prod[idx] + be[idx & (T_ * T_ - 1)];
  prod[idx] = 1.0f / (1.0f + __expf(-v));
}

// ---------------------------------------------------------------------------
// Kernel 5: E[b] = Ve (128x128) @ sig[b] (128x128)
// ---------------------------------------------------------------------------
__global__ __launch_bounds__(256) void k5_E(const float* __restrict__ Ve,
                                            const float* __restrict__ sig,
                                            float* __restrict__ E) {
  const int wid    = blockIdx.x * 8 + (threadIdx.x >> 5);   // 0..2047
  const int b      = wid >> 6;
  const int tile_m = (wid >> 3) & 7;
  const int tile_n = wid & 7;
  const int lane = threadIdx.x & 31;
  const int half = lane >> 4, lm = lane & 15;
  v8f c = wmma_tile_f32(Ve + (size_t)tile_m * 16 * T_, T_,
                        sig + (size_t)b * T_ * T_ + tile_n * 16, T_,
                        T_ / 4, lm, half);
  store_tile(E + (size_t)b * T_ * T_ + (size_t)tile_m * 16 * T_ + tile_n * 16,
             T_, c, lm, half);
}

// ---------------------------------------------------------------------------
// Kernel 6: out[b,t,r] = softmax over t of E[b,t,r]  (axis=1)
// one thread per (b,r) column; lanes cover consecutive r -> coalesced rows.
// Register-light 3-pass form: E (2 MB) and out (2 MB) are L2-resident, so
// re-reading beats spilling a 128-float per-thread buffer to scratch.
// ---------------------------------------------------------------------------
__global__ __launch_bounds__(256) void k6_softmax(const float* __restrict__ E,
                                                  float* __restrict__ out) {
  const int gid = blockIdx.x * 256 + threadIdx.x;   // < B*T
  const int b = gid >> 7;
  const int r = gid & (T_ - 1);
  const float* col = E + (size_t)b * T_ * T_ + r;
  float* ocol = out + (size_t)b * T_ * T_ + r;
  float m = -3.402823466e38f;
  for (int t = 0; t < T_; ++t) m = fmaxf(m, col[t * T_]);
  float s = 0.f;
  for (int t = 0; t < T_; ++t) {
    const float e = __expf(col[t * T_] - m);
    ocol[t * T_] = e;
    s += e;
  }
  const float inv = 1.0f / s;
  for (int t = 0; t < T_; ++t) ocol[t * T_] *= inv;
}

// ---------------------------------------------------------------------------
extern "C" void kernel_launch(void* const* d_in, const int* in_sizes, int n_in,
                              void* d_out, int out_size, void* d_ws, size_t ws_size,
                              hipStream_t stream) {
  const float* x  = (const float*)d_in[0];
  const float* U1 = (const float*)d_in[1];
  const float* U2 = (const float*)d_in[2];
  const float* U3 = (const float*)d_in[3];
  const float* be = (const float*)d_in[4];
  const float* Ve = (const float*)d_in[5];
  float* out = (float*)d_out;

  // workspace layout (floats): ~21 MB total
  float* ws       = (float*)d_ws;
  float* lhs_pre  = ws;                               // B*T*F   = 262144
  float* rhs      = lhs_pre + (size_t)B_ * T_ * F_;   // B*N*T   = 2097152
  float* lhs      = rhs + (size_t)B_ * N_ * T_;       // B*T*N   = 2097152
  float* prod     = lhs + (size_t)B_ * T_ * N_;       // B*T*T   = 524288 (-> sig)
  float* E        = prod + (size_t)B_ * T_ * T_;      // B*T*T   = 524288

  k0_zero<<<(B_ * T_ * F_ + 255) / 256, 256, 0, stream>>>(lhs_pre, B_ * T_ * F_);
  k1_stream<<<dim3(N_ / 32, B_), 256, 0, stream>>>(x, U1, U3, lhs_pre, rhs);
  k2_lhs<<<(B_ * T_ / 16) * (N_ / 16) / 8, 256, 0, stream>>>(lhs_pre, U2, lhs);
  k3_prod<<<B_ * (T_ / 16) * (T_ / 16) / 8, 256, 0, stream>>>(lhs, rhs, prod);
  k4_sig<<<B_ * T_ * T_ / 256, 256, 0, stream>>>(prod, be);
  k5_E<<<B_ * (T_ / 16) * (T_ / 16) / 8, 256, 0, stream>>>(Ve, prod, E);
  k6_softmax<<<B_ * T_ / 256, 256, 0, stream>>>(E, out);
}